// Embedding_10144712753720
// MI455X (gfx1250) — compile-verified
//
#include <hip/hip_runtime.h>

// Embedding: out[b,s,:] = content[token[b,s],:] + ((s-mean)/std) * pos[s,:]
// B=8, S=2048, V=50257, D=1024, all f32 except tokens.
// Pure bandwidth problem (~136 MiB moved -> ~6us floor at 23.3 TB/s).
// Strategy: 1 block per sequence position s; 256 threads x float4 = one
// 1024-wide row. Position row loaded once, scaled in registers, reused for
// all 8 batch rows. Content rows gathered with b128 loads (prefetched),
// output written with non-temporal b128 stores to keep L2 free for the
// ~196 MiB content table.
//
// Round 3 change: batch the 8 uniform token s_loads before any consumer so
// the compiler issues them back-to-back with ONE s_wait_kmcnt instead of 8
// serialized load->wait->use round trips (SMEM returns out of order; only
// kmcnt==0 waits exist).

#define BATCH 8
#define SEQ   2048
#define DIM   1024

// Native clang vector type: legal for __builtin_nontemporal_store and lowers
// to global_load_b128 / global_store_b128.
typedef float v4f __attribute__((ext_vector_type(4)));

__global__ __launch_bounds__(256) void embed_gather_addpos(
    const int*   __restrict__ tokens,   // [B, S]
    const float* __restrict__ content,  // [V, D]
    const float* __restrict__ pos,      // [S, D]
    float*       __restrict__ out)      // [B, S, D]
{
    const int s = blockIdx.x;           // sequence position, 0..2047
    const int d = threadIdx.x * 4;      // this thread's 4 floats of the dim

    // Phase 1: read the 8 tokens for this position (uniform per block ->
    // scalar loads). All independent: issue as one SMEM batch, one wait.
    int tok[BATCH];
#pragma unroll
    for (int b = 0; b < BATCH; ++b)
        tok[b] = tokens[b * SEQ + s];

    // Phase 2: kick off prefetch of each gathered content cacheline
    // (global_prefetch_b8) before any dependent use.
#pragma unroll
    for (int b = 0; b < BATCH; ++b)
        __builtin_prefetch(content + (size_t)tok[b] * DIM + d, /*rw=*/0, /*loc=*/1);

    // Population mean/std of t = 0..S-1 in closed form:
    //   mean = (S-1)/2,  var = (S^2-1)/12
    const float mean    = (float)(SEQ - 1) * 0.5f;
    const float var     = (float)((long long)SEQ * SEQ - 1) / 12.0f;
    const float inv_std = rsqrtf(var);
    const float scale   = ((float)s - mean) * inv_std;

    // Load + scale this thread's slice of the position row once; lives in
    // 4 VGPRs and is reused for all 8 batch entries.
    const v4f p  = *(const v4f*)(pos + (size_t)s * DIM + d);
    const v4f ps = p * scale;

    // Phase 3: gather + add + non-temporal streaming store.
#pragma unroll
    for (int b = 0; b < BATCH; ++b) {
        const v4f c = *(const v4f*)(content + (size_t)tok[b] * DIM + d);
        const v4f r = c + ps;
        v4f* dst = (v4f*)(out + ((size_t)b * SEQ + s) * DIM + d);
        __builtin_nontemporal_store(r, dst);
    }
}

extern "C" void kernel_launch(void* const* d_in, const int* in_sizes, int n_in,
                              void* d_out, int out_size, void* d_ws, size_t ws_size,
                              hipStream_t stream) {
    (void)in_sizes; (void)n_in; (void)out_size; (void)d_ws; (void)ws_size;
    const int*   tokens  = (const int*)  d_in[0];  // inputs [8,2048]
    const float* content = (const float*)d_in[1];  // content_kernel [50257,1024]
    const float* pos     = (const float*)d_in[2];  // position_kernel [2048,1024]
    float*       out     = (float*)      d_out;    // [8,2048,1024]

    dim3 grid(SEQ);    // 2048 blocks, one per sequence position
    dim3 block(256);   // 8 wave32 -> one 4KiB row per block via float4
    embed_gather_addpos<<<grid, block, 0, stream>>>(tokens, content, pos, out);
}